// ScoreBasedRecommendHead_17008070492355
// MI455X (gfx1250) — compile-verified
//
#include <hip/hip_runtime.h>
#include <hip/hip_bf16.h>

// ---------------------------------------------------------------------------
// ScoreBasedRecommendHead for MI455X (gfx1250):
//   scores = item_emb @ user_emb[user_id]  (GEMV, memory-bound: 512 MB stream,
//   roofline floor ~22 us @ 23.3 TB/s), then exact top-100 via histogram
//   threshold + candidate select (~8 MB extra traffic, negligible).
//
// CDNA5 paths used:
//   - global_load_async_to_lds_b128 (ASYNCcnt) : coalesced 512B/issue staging
//     of each wave's 16-item (8 KB contiguous) tile, th:TH_LOAD_NT so the
//     read-once 512 MB stream doesn't thrash the 192 MB L2.
//   - s_wait_asynccnt 0 before consuming the tile.
//   - v_wmma_f32_16x16x4_f32 : 32 WMMAs accumulate K=128 per 16-item tile,
//     user vector broadcast across B columns; D column 0 = scores (lanes 0/16).
// ---------------------------------------------------------------------------

#define EMB_D      128
#define NBUCKETS   4096
#define MAXCAND    4096
#define WAVES_PB   8          // waves per block
#define TILE_ITEMS 16         // items per wave

typedef float v2f __attribute__((ext_vector_type(2)));
typedef float v8f __attribute__((ext_vector_type(8)));

// raw LDS byte offset of a shared-memory pointer (ptrtoint of AS(3) pointer)
__device__ __forceinline__ unsigned lds_offset(const void* p) {
    return (unsigned)(size_t)(__attribute__((address_space(3))) const char*)p;
}

// ---------------- Kernel 1: async-staged WMMA GEMV scoring ------------------
// Block = 256 threads = 8 waves; each wave scores 16 items (8 KB tile).
__global__ void score_kernel(const int* __restrict__ user_id,
                             const float* __restrict__ user_emb,
                             const float* __restrict__ item_emb,
                             float* __restrict__ scores,
                             int n_items) {
    __shared__ float su[EMB_D];                              // user row (512 B)
    __shared__ float tiles[WAVES_PB * TILE_ITEMS * EMB_D];   // 64 KB, 8 KB/wave

    const int uid = *user_id;
    for (int i = threadIdx.x; i < EMB_D; i += blockDim.x)
        su[i] = user_emb[(long long)uid * EMB_D + i];
    __syncthreads();

    const int lane = threadIdx.x & 31;
    const int wave = threadIdx.x >> 5;
    const int half = (lane >> 4) & 1;   // 0: K pair {0,1}, 1: K pair {2,3}
    const int row  = lane & 15;         // item row within the 16-item tile

    const long long itemBase = ((long long)blockIdx.x * WAVES_PB + wave) * TILE_ITEMS;
    if (itemBase >= n_items) return;    // wave-uniform: EXEC stays all-ones

    // ---- stage this wave's 8 KB contiguous tile into its LDS region -------
    float* tile = &tiles[wave * TILE_ITEMS * EMB_D];
    const float* gbase = item_emb + itemBase * (long long)EMB_D;
    const unsigned ldsBase = lds_offset(tile);

#pragma unroll
    for (int it = 0; it < (TILE_ITEMS * EMB_D) / (32 * 4); ++it) {  // 16 issues
        const int chunk = it * 32 + lane;                 // 16 B chunk id
        const float* g  = gbase + chunk * 4;
        const unsigned loff = ldsBase + chunk * 16;
        // VDST = LDS byte offset, VADDR = 64-bit global address; NT: read-once
        asm volatile("global_load_async_to_lds_b128 %0, %1, off th:TH_LOAD_NT"
                     :: "v"(loff), "v"(g) : "memory");
    }
    asm volatile("s_wait_asynccnt 0x0" ::: "memory");

    // ---- 32 x V_WMMA_F32_16X16X4_F32 over K=128 ---------------------------
    // A frag (16x4 f32): lane<16 holds A[row][k0..k0+1], lane>=16 A[row][k0+2..3]
    const v2f* __restrict__ Arow = (const v2f*)(tile + row * EMB_D);
    const v2f* __restrict__ Urow = (const v2f*)su;

    v8f c = {};   // 16x16 f32 accumulator; all columns identical (B broadcast)
#pragma unroll
    for (int kc = 0; kc < EMB_D / 4; ++kc) {
        const int k2 = kc * 2 + half;       // float2 index = (kc*4 + 2*half)/2
        v2f a = Arow[k2];
        v2f b = Urow[k2];
        // 8 args: (neg_a, A, neg_b, B, c_mod, C, reuse_a, reuse_b)
        c = __builtin_amdgcn_wmma_f32_16x16x4_f32(
                false, a, false, b, (short)0, c, false, false);
    }

    // D column 0: lane 0 holds M=0..7 in c[0..7], lane 16 holds M=8..15.
    if (lane == 0 || lane == 16) {
        float* dst = scores + itemBase + (lane == 16 ? 8 : 0);
#pragma unroll
        for (int r = 0; r < 8; ++r) dst[r] = c[r];
    }
}

// ---------------- Kernel 2: zero histogram ----------------------------------
__global__ void zero_hist_kernel(unsigned* __restrict__ hist) {
    int i = blockIdx.x * blockDim.x + threadIdx.x;
    if (i < NBUCKETS) hist[i] = 0u;
}

// order-preserving transform: ascending float -> ascending uint
__device__ __forceinline__ unsigned order_key(float s) {
    unsigned u = __float_as_uint(s);
    return (u & 0x80000000u) ? ~u : (u | 0x80000000u);
}

// ---------------- Kernel 3: histogram of score keys -------------------------
__global__ void hist_kernel(const float* __restrict__ scores,
                            unsigned* __restrict__ hist, int n) {
    __shared__ unsigned lh[NBUCKETS];
    for (int i = threadIdx.x; i < NBUCKETS; i += blockDim.x) lh[i] = 0u;
    __syncthreads();

    const long long stride = (long long)gridDim.x * blockDim.x;
    for (long long i = (long long)blockIdx.x * blockDim.x + threadIdx.x;
         i < n; i += stride) {
        atomicAdd(&lh[order_key(scores[i]) >> 20], 1u);
    }
    __syncthreads();
    for (int i = threadIdx.x; i < NBUCKETS; i += blockDim.x)
        if (lh[i]) atomicAdd(&hist[i], lh[i]);
}

// ---------------- Kernel 4: suffix scan -> threshold bucket -----------------
// meta[0] = threshold bucket, meta[1] = candidate count (zeroed here)
__global__ void scan_kernel(const unsigned* __restrict__ hist,
                            unsigned* __restrict__ meta, int topk) {
    if (threadIdx.x == 0 && blockIdx.x == 0) {
        unsigned cum = 0;
        int b = NBUCKETS - 1;
        for (; b >= 0; --b) {
            cum += hist[b];
            if (cum >= (unsigned)topk) break;
        }
        meta[0] = (unsigned)(b < 0 ? 0 : b);
        meta[1] = 0u;
    }
}

// ---------------- Kernel 5: collect candidates ------------------------------
__global__ void collect_kernel(const float* __restrict__ scores,
                               unsigned* __restrict__ meta,
                               float* __restrict__ cScore,
                               int* __restrict__ cIdx, int n) {
    const unsigned thr = meta[0];
    const long long stride = (long long)gridDim.x * blockDim.x;
    for (long long i = (long long)blockIdx.x * blockDim.x + threadIdx.x;
         i < n; i += stride) {
        float s = scores[i];
        if ((order_key(s) >> 20) >= thr) {
            unsigned pos = atomicAdd(&meta[1], 1u);
            if (pos < MAXCAND) {
                cScore[pos] = s;
                cIdx[pos]   = (int)i;
            }
        }
    }
}

// ---------------- Kernel 6: select top-k from candidates --------------------
__global__ void select_kernel(const unsigned* __restrict__ meta,
                              float* __restrict__ cScore,
                              const int* __restrict__ cIdx,
                              int* __restrict__ out, int topk) {
    __shared__ float bs[256];
    __shared__ int   bp[256];

    const int M = (int)min(meta[1], (unsigned)MAXCAND);

    for (int t = 0; t < topk; ++t) {
        float best = -3.4e38f;
        int   pos  = -1;
        for (int i = threadIdx.x; i < M; i += blockDim.x) {
            float s = cScore[i];
            if (s > best) { best = s; pos = i; }
        }
        bs[threadIdx.x] = best;
        bp[threadIdx.x] = pos;
        __syncthreads();
        for (int s = 128; s > 0; s >>= 1) {
            if ((int)threadIdx.x < s) {
                if (bs[threadIdx.x + s] > bs[threadIdx.x]) {
                    bs[threadIdx.x] = bs[threadIdx.x + s];
                    bp[threadIdx.x] = bp[threadIdx.x + s];
                }
            }
            __syncthreads();
        }
        if (threadIdx.x == 0) {
            int p = bp[0];
            out[t] = (p >= 0) ? cIdx[p] : 0;
            if (p >= 0) cScore[p] = -3.4e38f;   // remove winner
        }
        __syncthreads();
    }
}

// ---------------------------------------------------------------------------
extern "C" void kernel_launch(void* const* d_in, const int* in_sizes, int n_in,
                              void* d_out, int out_size, void* d_ws, size_t ws_size,
                              hipStream_t stream) {
    const int*   user_id  = (const int*)d_in[0];
    const float* user_emb = (const float*)d_in[1];
    const float* item_emb = (const float*)d_in[2];
    (void)n_in; (void)ws_size;

    const int n_items = in_sizes[2] / EMB_D;
    const int topk    = out_size;

    // Workspace layout (n_items*4 is 256B-aligned for n_items=1e6)
    char* ws = (char*)d_ws;
    float*    scores = (float*)ws;
    size_t    off    = ((size_t)n_items * sizeof(float) + 255) & ~(size_t)255;
    unsigned* hist   = (unsigned*)(ws + off);          off += NBUCKETS * sizeof(unsigned);
    unsigned* meta   = (unsigned*)(ws + off);          off += 64;
    float*    cScore = (float*)(ws + off);             off += MAXCAND * sizeof(float);
    int*      cIdx   = (int*)(ws + off);

    int* out = (int*)d_out;   // reference output dtype is int32 indices

    // 1) GEMV scoring: 8 waves/block * 16 items/wave = 128 items per block
    const int scoreBlocks = (n_items + 127) / 128;
    score_kernel<<<scoreBlocks, 256, 0, stream>>>(user_id, user_emb, item_emb,
                                                  scores, n_items);

    // 2..6) exact top-k
    zero_hist_kernel<<<(NBUCKETS + 255) / 256, 256, 0, stream>>>(hist);
    hist_kernel<<<1024, 256, 0, stream>>>(scores, hist, n_items);
    scan_kernel<<<1, 32, 0, stream>>>(hist, meta, topk);
    collect_kernel<<<1024, 256, 0, stream>>>(scores, meta, cScore, cIdx, n_items);
    select_kernel<<<1, 256, 0, stream>>>(meta, cScore, cIdx, out, topk);
}